// RobertaPromptTune_73409581023290
// MI455X (gfx1250) — compile-verified
//
#include <hip/hip_runtime.h>
#include <hip/hip_bf16.h>
#include <math.h>

#define Bsz 32
#define Hn  16
#define Sn  128
#define Ln  3
#define Vn  50265

typedef __attribute__((ext_vector_type(2))) float v2f;
typedef __attribute__((ext_vector_type(8))) float v8f;

// ws layout (floats):
//   [0,96)      s[l*32+b]   (atomic accumulated)
//   [128,704)   anchor[l][b][j]  (3*32*6)
//   [1024]      loss_attn
//   [1025]      sum of per-layer supcon means
//   [1026]      loss_cls sum

__global__ void k_zero(float* ws) {
    int i = blockIdx.x * blockDim.x + threadIdx.x;
    if (i < 2048) ws[i] = 0.0f;
}

// One block (128 threads) per (l,b,h): masked sum of attention row q=mask_pos[b].
__global__ void k_attn_rows(const float* __restrict__ attn, const int* __restrict__ mask_pos,
                            const int* __restrict__ sep_pos, float* __restrict__ s_ws) {
    int blk = blockIdx.x;
    int h = blk % Hn;
    int b = (blk / Hn) % Bsz;
    int l = blk / (Hn * Bsz);
    int q  = mask_pos[b];
    int sp = sep_pos[b];
    int t = threadIdx.x;  // s index, 0..127
    size_t base = ((((size_t)l * Bsz + b) * Hn + h) * Sn + q) * (size_t)Sn;
    float v = attn[base + t];
    float in = (t >= q && t < sp - 1) ? v : 0.0f;
    __shared__ float sd[128];
    sd[t] = in;
    __syncthreads();
    for (int off = 64; off > 0; off >>= 1) {
        if (t < off) sd[t] += sd[t + off];
        __syncthreads();
    }
    if (t == 0) atomicAdd(&s_ws[l * Bsz + b], sd[0]);
}

// Sequential scans (96 + 3 steps) -> loss_attn. Single thread; trivially cheap.
__global__ void k_attn_scan(const float* __restrict__ s_ws, float* __restrict__ acc) {
    float T = 0.0f;
    float layer_T[3] = {0.0f, 0.0f, 0.0f};
    for (int idx = 0; idx < Ln * Bsz; ++idx) {
        T = (T + s_ws[idx]) / 16.0f;
        layer_T[idx / Bsz] += T;
    }
    float Ba = 0.0f, out = 0.0f;
    for (int l = 0; l < Ln; ++l) {
        Ba = (Ba + layer_T[l]) / 32.0f;
        out += Ba;
    }
    acc[0] = out / 3.0f;
}

// anchor[l,b,j] = mean over h of attention[l,b,h,6,j], j<6. One block per (l,b).
__global__ void k_anchor(const float* __restrict__ attn, float* __restrict__ anchor) {
    int blk = blockIdx.x;  // l*32 + b
    int b = blk % Bsz;
    int l = blk / Bsz;
    int t = threadIdx.x;   // 0..95
    __shared__ float sa[6];
    if (t < 6) sa[t] = 0.0f;
    __syncthreads();
    if (t < 96) {
        int j = t % 6, h = t / 6;
        size_t idx = (((size_t)l * Bsz + b) * Hn + h) * (size_t)(Sn * Sn) + 6 * Sn + j;
        atomicAdd(&sa[j], attn[idx]);
    }
    __syncthreads();
    if (t < 6) anchor[blk * 6 + t] = sa[t] / (float)Hn;
}

// One wave per layer. Gram G = a @ a^T via V_WMMA_F32_16X16X4_F32 (K=6 padded to 8),
// then supcon row math per lane. EXEC all-1s around the WMMA (no divergence).
__global__ void __launch_bounds__(32) k_supcon(const float* __restrict__ anchor,
                                               const int* __restrict__ labels,
                                               float* __restrict__ acc) {
    int l = blockIdx.x;
    int lane = threadIdx.x;  // 0..31
    __shared__ float aLds[32 * 8];  // a padded to K=8 with zeros
    __shared__ float G[32 * 32];
    __shared__ float norms[32];
    __shared__ int   larr[32];
    __shared__ float red[64];

    for (int j = 0; j < 6; ++j) aLds[lane * 8 + j] = anchor[(l * 32 + lane) * 6 + j];
    aLds[lane * 8 + 6] = 0.0f;
    aLds[lane * 8 + 7] = 0.0f;
    larr[lane] = labels[lane];
    __syncthreads();

    // A 16x4 f32 layout: VGPR0/1 = (K0,K1) for lanes 0-15, (K2,K3) for lanes 16-31.
    // B 4x16 mirrored; for a Gram matrix the per-lane A/B contents coincide per tile.
    int mrow = lane & 15;
    int grp  = lane >> 4;
    for (int mt = 0; mt < 2; ++mt) {
        for (int nt = 0; nt < 2; ++nt) {
            v8f c8 = {};
            for (int kc = 0; kc < 2; ++kc) {
                int k0 = 4 * kc + 2 * grp;
                v2f A, Bv;
                A.x  = aLds[(mt * 16 + mrow) * 8 + k0];
                A.y  = aLds[(mt * 16 + mrow) * 8 + k0 + 1];
                Bv.x = aLds[(nt * 16 + mrow) * 8 + k0];
                Bv.y = aLds[(nt * 16 + mrow) * 8 + k0 + 1];
                c8 = __builtin_amdgcn_wmma_f32_16x16x4_f32(
                        false, A, false, Bv, (short)0, c8, false, false);
            }
            // C/D layout: VGPR v holds M=v (lanes 0-15) / M=v+8 (lanes 16-31), N=lane&15
            for (int v = 0; v < 8; ++v) {
                int row = mt * 16 + v + (grp ? 8 : 0);
                int col = nt * 16 + mrow;
                G[row * 32 + col] = c8[v];
            }
        }
    }
    __syncthreads();

    norms[lane] = sqrtf(G[lane * 32 + lane]);
    __syncthreads();

    int li = larr[lane];
    float ni = norms[lane];
    float no_sim_sum = 0.0f;
    for (int j = 0; j < 32; ++j) {
        if (j == lane) continue;
        float denom = fmaxf(ni * norms[j], 1e-8f);
        float esim = expf(G[lane * 32 + j] / denom * 2.0f);  // sim/0.5
        if (li != larr[j]) no_sim_sum += esim;                // (esim - simm) row sum
    }
    float vsum = 0.0f;
    int vcnt = 0;
    for (int j = 0; j < 32; ++j) {
        float maskp = (li == larr[j]) ? 1.0f : 0.0f;
        float eye   = (j == lane) ? 1.0f : 0.0f;
        float esim  = (j == lane) ? 0.0f
                    : expf(G[lane * 32 + j] / fmaxf(ni * norms[j], 1e-8f) * 2.0f);
        float simm  = maskp * esim;
        float ratio = simm / (simm + no_sim_sum);
        float val   = -logf(1.0f - maskp + ratio + eye);
        vsum += val;
        vcnt += (val != 0.0f) ? 1 : 0;
    }
    red[lane] = vsum;
    red[32 + lane] = (float)vcnt;
    __syncthreads();
    if (lane == 0) {
        float s = 0.0f, c = 0.0f;
        for (int i = 0; i < 32; ++i) { s += red[i]; c += red[32 + i]; }
        atomicAdd(&acc[1], s / c);
    }
}

// One block per b: log-softmax over V of logits[b, mask_pos[b], :], then the
// tiny 8-token classifier head. This is the only bandwidth-significant kernel.
__global__ void k_cls(const float* __restrict__ logits, const float* __restrict__ position_w,
                      const float* __restrict__ pos_w, const float* __restrict__ neg_w,
                      const int* __restrict__ mask_pos, const int* __restrict__ labels,
                      const int* __restrict__ pos_ids, const int* __restrict__ neg_ids,
                      float* __restrict__ acc) {
    int b = blockIdx.x;
    int t = threadIdx.x;
    const float* row = logits + ((size_t)b * Sn + mask_pos[b]) * (size_t)Vn;
    __shared__ float red[256];

    float m = -INFINITY;
    for (int i = t; i < Vn; i += 256) m = fmaxf(m, row[i]);
    red[t] = m;
    __syncthreads();
    for (int off = 128; off > 0; off >>= 1) {
        if (t < off) red[t] = fmaxf(red[t], red[t + off]);
        __syncthreads();
    }
    float maxv = red[0];
    __syncthreads();

    float s = 0.0f;
    for (int i = t; i < Vn; i += 256) s += expf(row[i] - maxv);
    red[t] = s;
    __syncthreads();
    for (int off = 128; off > 0; off >>= 1) {
        if (t < off) red[t] += red[t + off];
        __syncthreads();
    }

    if (t == 0) {
        float logZ = maxv + logf(red[0]);
        float p0 = position_w[0];
        float pwv[4], nwv[4];
        float pm = -INFINITY, nm = -INFINITY;
        for (int k = 0; k < 4; ++k) {
            pwv[k] = pos_w[k]; pm = fmaxf(pm, pwv[k]);
            nwv[k] = neg_w[k]; nm = fmaxf(nm, nwv[k]);
        }
        float ps = 0.0f, ns = 0.0f;
        for (int k = 0; k < 4; ++k) {
            pwv[k] = expf(pwv[k] - pm); ps += pwv[k];
            nwv[k] = expf(nwv[k] - nm); ns += nwv[k];
        }
        float pos = 0.0f, neg = 0.0f;
        for (int k = 0; k < 4; ++k) {
            pos += (pwv[k] / ps) * ((row[pos_ids[k]] - logZ) * p0);
            neg += (nwv[k] / ns) * ((row[neg_ids[k]] - logZ) * p0);
        }
        // cls = softmax([neg, pos]); logp = log_softmax(cls); pick labels[b]
        float mm = fmaxf(neg, pos);
        float e0 = expf(neg - mm), e1 = expf(pos - mm);
        float c0 = e0 / (e0 + e1), c1 = e1 / (e0 + e1);
        float mm2 = fmaxf(c0, c1);
        float lse = mm2 + logf(expf(c0 - mm2) + expf(c1 - mm2));
        float sel = labels[b] ? c1 : c0;
        atomicAdd(&acc[2], -(sel - lse) / (float)Bsz);
    }
}

__global__ void k_combine(const float* __restrict__ acc, float* __restrict__ out) {
    out[0] = acc[2] + 0.3f * (acc[1] / 3.0f) + 0.3f * acc[0];
}

extern "C" void kernel_launch(void* const* d_in, const int* in_sizes, int n_in,
                              void* d_out, int out_size, void* d_ws, size_t ws_size,
                              hipStream_t stream) {
    const float* attention  = (const float*)d_in[0];
    const float* logits     = (const float*)d_in[1];
    const float* position_w = (const float*)d_in[2];
    const float* pos_w      = (const float*)d_in[3];
    const float* neg_w      = (const float*)d_in[4];
    const int*   mask_pos   = (const int*)d_in[5];
    const int*   sep_pos    = (const int*)d_in[6];
    const int*   labels     = (const int*)d_in[7];
    const int*   pos_ids    = (const int*)d_in[8];
    const int*   neg_ids    = (const int*)d_in[9];

    float* ws     = (float*)d_ws;
    float* s_ws   = ws;          // 96 floats
    float* anchor = ws + 128;    // 576 floats
    float* acc    = ws + 1024;   // 3 floats
    float* out    = (float*)d_out;

    hipLaunchKernelGGL(k_zero,      dim3(8),              dim3(256), 0, stream, ws);
    hipLaunchKernelGGL(k_attn_rows, dim3(Ln * Bsz * Hn),  dim3(128), 0, stream,
                       attention, mask_pos, sep_pos, s_ws);
    hipLaunchKernelGGL(k_anchor,    dim3(Ln * Bsz),       dim3(96),  0, stream,
                       attention, anchor);
    hipLaunchKernelGGL(k_attn_scan, dim3(1),              dim3(1),   0, stream, s_ws, acc);
    hipLaunchKernelGGL(k_supcon,    dim3(Ln),             dim3(32),  0, stream,
                       anchor, labels, acc);
    hipLaunchKernelGGL(k_cls,       dim3(Bsz),            dim3(256), 0, stream,
                       logits, position_w, pos_w, neg_w, mask_pos, labels,
                       pos_ids, neg_ids, acc);
    hipLaunchKernelGGL(k_combine,   dim3(1),              dim3(1),   0, stream, acc, out);
}